// AttnSenseNet_30090540876285
// MI455X (gfx1250) — compile-verified
//
#include <hip/hip_runtime.h>
#include <hip/hip_bf16.h>
#include <math.h>

// Problem constants (from reference)
#define BATCH     64
#define LWORDS    512
#define NSENSE    3
#define DIM       128
#define OUTV      50000
#define LSTOT     (LWORDS * NSENSE)   // 1536

typedef __attribute__((ext_vector_type(2))) float v2f;
typedef __attribute__((ext_vector_type(8))) float v8f;

// ---------------------------------------------------------------------------
// wave32 butterfly all-reduce (sum)
__device__ __forceinline__ float wave_all_sum(float v) {
    v += __shfl_xor(v, 16, 32);
    v += __shfl_xor(v, 8, 32);
    v += __shfl_xor(v, 4, 32);
    v += __shfl_xor(v, 2, 32);
    v += __shfl_xor(v, 1, 32);
    return v;
}

// ---------------------------------------------------------------------------
// Stage 1: word_imp[b,l] = mask ? -inf : dot(mean_sense_embed, W_attn) + b_attn
// One wave per (b,l) word. Lane owns dims d = lane + 32*j, j=0..3 (coalesced).
__global__ __launch_bounds__(256) void k_word_imp(
    const int* __restrict__ inputs,          // [B, L*S]
    const unsigned char* __restrict__ mask,  // [B, L] (bool as byte)
    const float* __restrict__ embedding,     // [V, D]
    const float* __restrict__ W_attn,        // [1, D]
    const float* __restrict__ b_attn,        // [1]
    float* __restrict__ word_imp)            // [B, L] ws
{
    const int wv   = threadIdx.x >> 5;
    const int lane = threadIdx.x & 31;
    const int word = blockIdx.x * 8 + wv;            // global word id
    if (word >= BATCH * LWORDS) return;
    const int b = word / LWORDS;
    const int l = word - b * LWORDS;

    float mean[4] = {0.f, 0.f, 0.f, 0.f};
    #pragma unroll
    for (int s = 0; s < NSENSE; ++s) {
        const int tok = inputs[b * LSTOT + l * NSENSE + s];
        if (tok != 0) {                               // PAD row contributes 0
            const float* er = embedding + (size_t)tok * DIM;
            #pragma unroll
            for (int j = 0; j < 4; ++j) mean[j] += er[lane + 32 * j];
        }
    }
    float p = 0.f;
    #pragma unroll
    for (int j = 0; j < 4; ++j) {
        mean[j] *= (1.0f / NSENSE);
        p += mean[j] * W_attn[lane + 32 * j];
    }
    p = wave_all_sum(p);
    if (lane == 0) {
        float imp = p + b_attn[0];
        if (mask[b * LWORDS + l]) imp = -__builtin_inff();
        word_imp[b * LWORDS + l] = imp;
    }
}

// ---------------------------------------------------------------------------
// Stage 2: word softmax over L, context[b,:] = sum_l w_l * embed_mean[b,l,:]
// One block (256 threads = 8 waves) per batch row. Means re-gathered (L2-hot).
__global__ __launch_bounds__(256) void k_context(
    const int* __restrict__ inputs,
    const float* __restrict__ embedding,
    const float* __restrict__ word_imp,   // [B, L]
    float* __restrict__ context)          // [B, D] ws
{
    __shared__ float red[256];
    __shared__ float ctx[DIM];
    __shared__ float s_max, s_inv;

    const int b    = blockIdx.x;
    const int tid  = threadIdx.x;
    const int wv   = tid >> 5;
    const int lane = tid & 31;
    const float* wi = word_imp + b * LWORDS;

    // block max over 512
    float m = fmaxf(wi[2 * tid], wi[2 * tid + 1]);
    red[tid] = m;  __syncthreads();
    for (int s = 128; s > 0; s >>= 1) {
        if (tid < s) red[tid] = fmaxf(red[tid], red[tid + s]);
        __syncthreads();
    }
    if (tid == 0) s_max = red[0];
    __syncthreads();
    const float mx = s_max;

    // block sum of exp
    float sm = expf(wi[2 * tid] - mx) + expf(wi[2 * tid + 1] - mx);
    red[tid] = sm;  __syncthreads();
    for (int s = 128; s > 0; s >>= 1) {
        if (tid < s) red[tid] += red[tid + s];
        __syncthreads();
    }
    if (tid == 0) s_inv = 1.0f / red[0];
    if (tid < DIM) ctx[tid] = 0.f;
    __syncthreads();
    const float inv = s_inv;

    // weighted accumulation of means; wave wv owns words l = wv, wv+8, ...
    float acc[4] = {0.f, 0.f, 0.f, 0.f};
    for (int l = wv; l < LWORDS; l += 8) {
        const float w = expf(wi[l] - mx) * inv;
        float mean[4] = {0.f, 0.f, 0.f, 0.f};
        #pragma unroll
        for (int s = 0; s < NSENSE; ++s) {
            const int tok = inputs[b * LSTOT + l * NSENSE + s];
            if (tok != 0) {
                const float* er = embedding + (size_t)tok * DIM;
                #pragma unroll
                for (int j = 0; j < 4; ++j) mean[j] += er[lane + 32 * j];
            }
        }
        #pragma unroll
        for (int j = 0; j < 4; ++j) acc[j] += w * (mean[j] * (1.0f / NSENSE));
    }
    #pragma unroll
    for (int j = 0; j < 4; ++j) atomicAdd(&ctx[lane + 32 * j], acc[j]);
    __syncthreads();
    if (tid < DIM) context[b * DIM + tid] = ctx[tid];
}

// ---------------------------------------------------------------------------
// Stage 3: sense attention -> hidden[b,:]
__global__ __launch_bounds__(256) void k_hidden(
    const int* __restrict__ inputs,
    const float* __restrict__ length_weights,  // [B]
    const float* __restrict__ embedding,
    const float* __restrict__ context,         // [B, D]
    float* __restrict__ hidden)                // [B, D] ws
{
    __shared__ float hid[DIM];
    const int b    = blockIdx.x;
    const int tid  = threadIdx.x;
    const int wv   = tid >> 5;
    const int lane = tid & 31;

    float cx[4];
    #pragma unroll
    for (int j = 0; j < 4; ++j) cx[j] = context[b * DIM + lane + 32 * j];
    const float lw = length_weights[b];

    if (tid < DIM) hid[tid] = 0.f;
    __syncthreads();

    float acc[4] = {0.f, 0.f, 0.f, 0.f};
    for (int l = wv; l < LWORDS; l += 8) {
        float e[NSENSE][4];
        float sim[NSENSE];
        int   tok[NSENSE];
        #pragma unroll
        for (int s = 0; s < NSENSE; ++s) {
            tok[s] = inputs[b * LSTOT + l * NSENSE + s];
            float p = 0.f;
            if (tok[s] != 0) {
                const float* er = embedding + (size_t)tok[s] * DIM;
                #pragma unroll
                for (int j = 0; j < 4; ++j) { e[s][j] = er[lane + 32 * j]; p += e[s][j] * cx[j]; }
            } else {
                #pragma unroll
                for (int j = 0; j < 4; ++j) e[s][j] = 0.f;
            }
            sim[s] = wave_all_sum(p);   // all lanes hold the dot product
        }
        const float mxs = fmaxf(sim[0], fmaxf(sim[1], sim[2]));
        float es[NSENSE], ssum = 0.f;
        #pragma unroll
        for (int s = 0; s < NSENSE; ++s) { es[s] = expf(sim[s] - mxs); ssum += es[s]; }
        const float sc = lw / ssum;
        #pragma unroll
        for (int s = 0; s < NSENSE; ++s) {
            const float w = (tok[s] == 0) ? 0.f : es[s] * sc;
            #pragma unroll
            for (int j = 0; j < 4; ++j) acc[j] += w * e[s][j];
        }
    }
    #pragma unroll
    for (int j = 0; j < 4; ++j) atomicAdd(&hid[lane + 32 * j], acc[j]);
    __syncthreads();
    if (tid < DIM) hidden[b * DIM + tid] = hid[tid];
}

// ---------------------------------------------------------------------------
// Stage 4: logits = hidden[64,128] @ W_lin^T + b_lin  via V_WMMA_F32_16X16X4_F32
// Block: 128 threads (4 waves), covers 64 output columns x all 64 rows.
// LDS: A = hidden (64x128, stride 132), B = W_lin tile (64 rows x 128, stride 132).
#define LDSTRIDE 132
__global__ __launch_bounds__(128) void k_gemm_wmma(
    const float* __restrict__ hidden,   // [64, 128]
    const float* __restrict__ W_lin,    // [OUTV, 128]
    const float* __restrict__ b_lin,    // [OUTV]
    float* __restrict__ out)            // [64, OUTV] (raw logits)
{
    __shared__ float hA[64 * LDSTRIDE];
    __shared__ float hB[64 * LDSTRIDE];

    const int tid  = threadIdx.x;
    const int wv   = tid >> 5;
    const int lane = tid & 31;
    const int n0   = blockIdx.x * 64;

    // Stage hidden and W_lin tile into LDS (coalesced over the 128-dim).
    for (int idx = tid; idx < 64 * DIM; idx += 128) {
        const int r = idx >> 7, c = idx & 127;
        hA[r * LDSTRIDE + c] = hidden[idx];
        const int gr = n0 + r;
        hB[r * LDSTRIDE + c] = (gr < OUTV) ? W_lin[(size_t)gr * DIM + c] : 0.f;
    }
    __syncthreads();

    // Per ISA 7.12.2 (32-bit 16x4 A / 4x16 B): lanes 0-15 -> K{0,1}, 16-31 -> K{2,3}
    const int nl    = lane & 15;            // column within 16-wide N tile
    const int khalf = (lane >> 4) * 2;      // 0 or 2
    const int cn    = wv * 16;              // this wave's N tile inside block
    const int rowhi = (lane >> 4) << 3;     // +8 rows for upper half lanes

    const int   gcol  = n0 + cn + nl;
    const bool  valid = (gcol < OUTV);
    const float bias  = valid ? b_lin[gcol] : 0.f;
    const float* bcol = &hB[(cn + nl) * LDSTRIDE];

    #pragma unroll
    for (int mt = 0; mt < 4; ++mt) {
        const int m0 = mt * 16;
        const float* arow = &hA[(m0 + nl) * LDSTRIDE];
        v8f c = {0.f, 0.f, 0.f, 0.f, 0.f, 0.f, 0.f, 0.f};
        #pragma unroll
        for (int kk = 0; kk < 32; ++kk) {
            const int kb = kk * 4 + khalf;
            v2f a = {arow[kb], arow[kb + 1]};
            v2f bb = {bcol[kb], bcol[kb + 1]};
            c = __builtin_amdgcn_wmma_f32_16x16x4_f32(
                    /*neg_a=*/false, a, /*neg_b=*/false, bb,
                    /*c_mod=*/(short)0, c, /*reuse_a=*/false, /*reuse_b=*/false);
        }
        if (valid) {
            #pragma unroll
            for (int i = 0; i < 8; ++i)
                out[(size_t)(m0 + rowhi + i) * OUTV + gcol] = c[i] + bias;
        }
    }
}

// ---------------------------------------------------------------------------
// Stage 5: in-place log_softmax per row of d_out [64, OUTV]
__global__ __launch_bounds__(256) void k_logsoftmax(float* __restrict__ out)
{
    __shared__ float red[256];
    __shared__ float s_lse;
    const int tid = threadIdx.x;
    float* row = out + (size_t)blockIdx.x * OUTV;

    float m = -__builtin_inff();
    for (int i = tid; i < OUTV; i += 256) m = fmaxf(m, row[i]);
    red[tid] = m;  __syncthreads();
    for (int s = 128; s > 0; s >>= 1) {
        if (tid < s) red[tid] = fmaxf(red[tid], red[tid + s]);
        __syncthreads();
    }
    const float mx = red[0];
    __syncthreads();

    float sm = 0.f;
    for (int i = tid; i < OUTV; i += 256) sm += expf(row[i] - mx);
    red[tid] = sm;  __syncthreads();
    for (int s = 128; s > 0; s >>= 1) {
        if (tid < s) red[tid] += red[tid + s];
        __syncthreads();
    }
    if (tid == 0) s_lse = mx + logf(red[0]);
    __syncthreads();
    const float lse = s_lse;

    for (int i = tid; i < OUTV; i += 256) row[i] = row[i] - lse;
}

// ---------------------------------------------------------------------------
extern "C" void kernel_launch(void* const* d_in, const int* in_sizes, int n_in,
                              void* d_out, int out_size, void* d_ws, size_t ws_size,
                              hipStream_t stream) {
    const int*           inputs   = (const int*)d_in[0];
    const float*         len_w    = (const float*)d_in[1];
    const unsigned char* mask     = (const unsigned char*)d_in[2]; // bool mask
    const float*         embed    = (const float*)d_in[3];
    const float*         W_attn   = (const float*)d_in[4];
    const float*         b_attn   = (const float*)d_in[5];
    const float*         W_lin    = (const float*)d_in[6];
    const float*         b_lin    = (const float*)d_in[7];
    float*               out      = (float*)d_out;

    // workspace layout (floats)
    float* ws        = (float*)d_ws;
    float* word_imp  = ws;                          // B*L      = 32768
    float* context   = word_imp + BATCH * LWORDS;   // B*D      =  8192
    float* hidden    = context  + BATCH * DIM;      // B*D      =  8192

    // 1) word importance: one wave per word, 8 waves/block
    k_word_imp<<<(BATCH * LWORDS) / 8, 256, 0, stream>>>(
        inputs, mask, embed, W_attn, b_attn, word_imp);

    // 2) word softmax + context
    k_context<<<BATCH, 256, 0, stream>>>(inputs, embed, word_imp, context);

    // 3) sense attention + hidden
    k_hidden<<<BATCH, 256, 0, stream>>>(inputs, len_w, embed, context, hidden);

    // 4) WMMA GEMM -> raw logits in d_out
    k_gemm_wmma<<<(OUTV + 63) / 64, 128, 0, stream>>>(hidden, W_lin, b_lin, out);

    // 5) in-place log_softmax per row
    k_logsoftmax<<<BATCH, 256, 0, stream>>>(out);
}